// GCN_3Layers_21388937134410
// MI455X (gfx1250) — compile-verified
//
#include <hip/hip_runtime.h>
#include <hip/hip_bf16.h>

typedef float v2f __attribute__((ext_vector_type(2)));
typedef float v8f __attribute__((ext_vector_type(8)));

#define N_NODES 100000

// ---------------------------------------------------------------------------
// Degree accumulation: one thread per edge, float atomics into deg arrays.
// ---------------------------------------------------------------------------
__global__ void deg_kernel(const int* __restrict__ src, const int* __restrict__ dst,
                           float* __restrict__ deg_out, float* __restrict__ deg_in, int E) {
    int e = blockIdx.x * blockDim.x + threadIdx.x;
    if (e < E) {
        atomicAdd(&deg_out[src[e]], 1.0f);
        atomicAdd(&deg_in[dst[e]], 1.0f);
    }
}

// In-place deg -> rsqrt(max(deg,1))
__global__ void norm_kernel(float* __restrict__ deg_out, float* __restrict__ deg_in, int n) {
    int i = blockIdx.x * blockDim.x + threadIdx.x;
    if (i < n) {
        float o = deg_out[i];
        float d = deg_in[i];
        deg_out[i] = rsqrtf(o < 1.0f ? 1.0f : o);
        deg_in[i]  = rsqrtf(d < 1.0f ? 1.0f : d);
    }
}

// ---------------------------------------------------------------------------
// Dense GEMM Y[M,N] = X[M,K] @ W[K,N] using V_WMMA_F32_16X16X4_F32.
// Each wave owns MT=2 row tiles (32 rows, M must be a multiple of 32) and all
// N/16 column tiles. Sharing each B fragment across 2 WMMAs halves B-load
// traffic per matrix op vs. the single-tile version (1 b32 load per WMMA).
// A fragment (16x4 f32): lanes 0-15 -> K = k+0,k+1 ; lanes 16-31 -> K = k+2,k+3
// B fragment (4x16 f32): lane half selects K rows, lane&15 selects column.
// C/D (16x16 f32): VGPR j = row j (lanes 0-15) / row j+8 (lanes 16-31).
// ---------------------------------------------------------------------------
template <int K, int N>
__global__ __launch_bounds__(256) void gemm_wmma_f32(const float* __restrict__ X,
                                                     const float* __restrict__ W,
                                                     float* __restrict__ Y, int M) {
    constexpr int NT = N / 16;
    const int wave = threadIdx.x >> 5;
    const int lane = threadIdx.x & 31;
    const int job  = blockIdx.x * (blockDim.x >> 5) + wave;
    const int row0 = job * 32;
    if (row0 >= M) return;  // wave-uniform: EXEC stays all-ones for WMMA

    const int half = lane >> 4;   // 0: K lanes 0/1, 1: K lanes 2/3
    const int l    = lane & 15;   // row (A) / column (B,C,D) within tile

    const float* __restrict__ xp0 = X + (size_t)(row0 + l) * K + half * 2;
    const float* __restrict__ xp1 = xp0 + (size_t)16 * K;

    v8f acc0[NT];
    v8f acc1[NT];
#pragma unroll
    for (int n = 0; n < NT; ++n) { acc0[n] = v8f{}; acc1[n] = v8f{}; }

    for (int kk = 0; kk < K; kk += 4) {
        v2f a0, a1;
        a0.x = xp0[kk];
        a0.y = xp0[kk + 1];
        a1.x = xp1[kk];
        a1.y = xp1[kk + 1];
        const int kr = kk + half * 2;
        const float* __restrict__ wp0 = W + (size_t)kr * N + l;
        const float* __restrict__ wp1 = wp0 + N;
#pragma unroll
        for (int n = 0; n < NT; ++n) {
            v2f b;
            b.x = wp0[n * 16];
            b.y = wp1[n * 16];
            acc0[n] = __builtin_amdgcn_wmma_f32_16x16x4_f32(
                false, a0, false, b, (short)0, acc0[n], false, false);
            acc1[n] = __builtin_amdgcn_wmma_f32_16x16x4_f32(
                false, a1, false, b, (short)0, acc1[n], false, false);
        }
    }

#pragma unroll
    for (int n = 0; n < NT; ++n) {
        const int col = n * 16 + l;
#pragma unroll
        for (int j = 0; j < 8; ++j) {
            const int r = j + half * 8;
            Y[(size_t)(row0 + r) * N + col]      = acc0[n][j];
            Y[(size_t)(row0 + 16 + r) * N + col] = acc1[n][j];
        }
    }
}

// ---------------------------------------------------------------------------
// Edge scatter: one wave per edge; lane handles D/32 contiguous features.
// agg[dst] += xw[src] * inv_sqrt_out[src]
// ---------------------------------------------------------------------------
template <int D>
__global__ void scatter_kernel(const float* __restrict__ xw,
                               const int* __restrict__ src, const int* __restrict__ dst,
                               const float* __restrict__ inv_o,
                               float* __restrict__ agg, int E) {
    constexpr int PER = D / 32;
    int gid  = blockIdx.x * blockDim.x + threadIdx.x;
    int e    = gid >> 5;
    int lane = gid & 31;
    if (e >= E) return;
    int s = src[e];
    int d = dst[e];
    float w = inv_o[s];
    const float* __restrict__ xs = xw + (size_t)s * D + lane * PER;
    float* __restrict__ op = agg + (size_t)d * D + lane * PER;
    float vals[PER];
#pragma unroll
    for (int i = 0; i < PER; ++i) vals[i] = xs[i];
#pragma unroll
    for (int i = 0; i < PER; ++i) atomicAdd(op + i, vals[i] * w);
}

// ---------------------------------------------------------------------------
// Finalize: out = [relu](agg * inv_sqrt_in + bias)
// ---------------------------------------------------------------------------
template <int D, bool RELU>
__global__ void finalize_kernel(const float* __restrict__ agg,
                                const float* __restrict__ inv_i,
                                const float* __restrict__ bias,
                                float* __restrict__ out, int nNodes) {
    int i = blockIdx.x * blockDim.x + threadIdx.x;
    if (i >= nNodes * D) return;
    int node = i / D;
    int f    = i & (D - 1);
    float v = agg[i] * inv_i[node] + bias[f];
    if (RELU) v = fmaxf(v, 0.0f);
    out[i] = v;
}

extern "C" void kernel_launch(void* const* d_in, const int* in_sizes, int n_in,
                              void* d_out, int out_size, void* d_ws, size_t ws_size,
                              hipStream_t stream) {
    (void)n_in; (void)out_size; (void)ws_size;
    const float* x   = (const float*)d_in[0];
    const float* W1  = (const float*)d_in[1];
    const float* b1  = (const float*)d_in[2];
    const float* W2  = (const float*)d_in[3];
    const float* b2  = (const float*)d_in[4];
    const float* W3  = (const float*)d_in[5];
    const float* b3  = (const float*)d_in[6];
    const int*   src = (const int*)d_in[7];
    const int*   dst = (const int*)d_in[8];
    const int E = in_sizes[7];
    float* out = (float*)d_out;

    // Workspace layout
    float* ws    = (float*)d_ws;
    float* inv_o = ws;
    float* inv_i = inv_o + N_NODES;
    float* bufA  = inv_i + N_NODES;                      // GEMM output  [N,128]
    float* bufB  = bufA + (size_t)N_NODES * 128;         // aggregation  [N,128]
    float* bufC  = bufB + (size_t)N_NODES * 128;         // activations  [N,128]

    const int jobs       = N_NODES / 32;                 // 3125 (exact, M % 32 == 0)
    const int gemmBlocks = (jobs + 7) / 8;               // 8 waves / block
    const int edgeThreads = E * 32;
    const int edgeBlocks  = (edgeThreads + 255) / 256;
    const int degBlocks   = (E + 255) / 256;
    const int nodeBlocks  = (N_NODES + 255) / 256;
    const int f128Blocks  = (N_NODES * 128 + 255) / 256;
    const int f64Blocks   = (N_NODES * 64 + 255) / 256;

    // ---- degree norms ----
    hipMemsetAsync(inv_o, 0, 2 * (size_t)N_NODES * sizeof(float), stream);
    deg_kernel<<<degBlocks, 256, 0, stream>>>(src, dst, inv_o, inv_i, E);
    norm_kernel<<<nodeBlocks, 256, 0, stream>>>(inv_o, inv_i, N_NODES);

    // ---- layer 1: (agg is linear) h1 = relu(agg(x @ W1) + b1) ----
    gemm_wmma_f32<256, 128><<<gemmBlocks, 256, 0, stream>>>(x, W1, bufA, N_NODES);
    hipMemsetAsync(bufB, 0, (size_t)N_NODES * 128 * sizeof(float), stream);
    scatter_kernel<128><<<edgeBlocks, 256, 0, stream>>>(bufA, src, dst, inv_o, bufB, E);
    finalize_kernel<128, true><<<f128Blocks, 256, 0, stream>>>(bufB, inv_i, b1, bufC, N_NODES);

    // ---- layer 2 ----
    gemm_wmma_f32<128, 128><<<gemmBlocks, 256, 0, stream>>>(bufC, W2, bufA, N_NODES);
    hipMemsetAsync(bufB, 0, (size_t)N_NODES * 128 * sizeof(float), stream);
    scatter_kernel<128><<<edgeBlocks, 256, 0, stream>>>(bufA, src, dst, inv_o, bufB, E);
    finalize_kernel<128, true><<<f128Blocks, 256, 0, stream>>>(bufB, inv_i, b2, bufC, N_NODES);

    // ---- layer 3 (no relu) -> d_out ----
    gemm_wmma_f32<128, 64><<<gemmBlocks, 256, 0, stream>>>(bufC, W3, bufA, N_NODES);
    hipMemsetAsync(bufB, 0, (size_t)N_NODES * 64 * sizeof(float), stream);
    scatter_kernel<64><<<edgeBlocks, 256, 0, stream>>>(bufA, src, dst, inv_o, bufB, E);
    finalize_kernel<64, false><<<f64Blocks, 256, 0, stream>>>(bufB, inv_i, b3, out, N_NODES);
}